// SkeletalPool_17798344475160
// MI455X (gfx1250) — compile-verified
//
#include <hip/hip_runtime.h>

// SkeletalPool: out[b, j, c, t] = 0.5 * (x[b, i0(j), c, t] + x[b, i1(j), c, t])
// x : [32, 31, 64, 4096] f32   (plane = 64*4096 floats = 65536 float4 = 1 MB)
// out: [32, 16, 64, 4096] f32
//
// Memory-bound streaming kernel (AI ~= 0.17 FLOP/byte): optimized for the
// 23.3 TB/s HBM path with 128-bit global ops + non-temporal hints (the 1.6 GB
// stream does not fit the 192 MB L2, so NT/write-rinse is the right policy).

typedef __attribute__((ext_vector_type(4))) float v4f;

__global__ __launch_bounds__(256) void skel_pool_b128_nt(
    const float* __restrict__ xf, float* __restrict__ outf) {
  const v4f* __restrict__ x = (const v4f*)xf;
  v4f* __restrict__ o = (v4f*)outf;

  // 2^22 threads total. Thread id decomposes as {b0[1:0], j[3:0], off[15:0]}
  // in float4 units; the per-thread stride of 2^22 float4 advances b by 4,
  // keeping j/off/i0/i1 loop-invariant.
  const unsigned t   = blockIdx.x * 256u + threadIdx.x;  // < 2^22
  const unsigned off = t & 65535u;        // float4 offset within a plane
  const unsigned j   = (t >> 16) & 15u;   // output joint 0..15
  const unsigned b0  = t >> 20;           // 0..3
  const unsigned i1  = 2u * j;            // second pooled input joint
  const unsigned i0  = j ? (i1 - 1u) : 0u;

#pragma unroll
  for (unsigned k = 0; k < 8u; ++k) {
    const unsigned b   = b0 + 4u * k;      // batch 0..31
    const unsigned pin = b * 31u;          // input plane base for this batch
    // Two 128-bit non-temporal loads (each input plane is consumed once).
    v4f a = __builtin_nontemporal_load(&x[((pin + i0) << 16) + off]);
    v4f c = __builtin_nontemporal_load(&x[((pin + i1) << 16) + off]);
    v4f r = (a + c) * 0.5f;
    // 128-bit non-temporal store; out idx = ((b*16 + j) << 16) + off
    __builtin_nontemporal_store(r, &o[(((b << 4) + j) << 16) + off]);
  }
}

extern "C" void kernel_launch(void* const* d_in, const int* in_sizes, int n_in,
                              void* d_out, int out_size, void* d_ws, size_t ws_size,
                              hipStream_t stream) {
  (void)in_sizes; (void)n_in; (void)d_ws; (void)ws_size; (void)out_size;
  const float* x = (const float*)d_in[0];   // [32,31,64,4096] f32
  float* out = (float*)d_out;               // [32,16,64,4096] f32

  // Total output: 32*16*65536 = 2^25 float4; 8 per thread -> 2^22 threads.
  skel_pool_b128_nt<<<dim3(16384), dim3(256), 0, stream>>>(x, out);
}